// BuildnetEnc_Edge_31550829756489
// MI455X (gfx1250) — compile-verified
//
#include <hip/hip_runtime.h>
#include <hip/hip_bf16.h>

// ---------------------------------------------------------------------------
// GNN encoder for MI455X (gfx1250, wave32, WMMA).
// All GEMMs run on V_WMMA_F32_16X16X4_F32 (full fp32 precision — problem is
// HBM-bound at ~1.7GB traffic => ~75us floor @ 23.3TB/s, compute is cheap).
// Weights are repacked once per launch into WMMA-B-fragment order so each
// B fragment is one global_load_b64; the indexed gather goes through the
// CDNA5 async-to-LDS path (ASYNCcnt) when the builtins are available.
// ---------------------------------------------------------------------------

#define GN_N 25000
#define GN_E 400000
#define GN_DN 128
#define GN_DE 64

typedef __attribute__((ext_vector_type(2))) float v2f;
typedef __attribute__((ext_vector_type(8))) float v8f;
typedef __attribute__((ext_vector_type(4))) int v4i;

#if __has_builtin(__builtin_amdgcn_global_load_async_to_lds_b128) && \
    __has_builtin(__builtin_amdgcn_s_wait_asynccnt)
#define GN_ASYNC 1
typedef __attribute__((address_space(1))) v4i gv4i;  // global int4 ("__device__")
typedef __attribute__((address_space(3))) v4i lv4i;  // LDS int4 ("__shared__")
#else
#define GN_ASYNC 0
#endif

// Copy 16 bytes global -> LDS (async path if available).
__device__ __forceinline__ void copy16_to_lds(float* lds_dst, const float* gsrc) {
#if GN_ASYNC
  __builtin_amdgcn_global_load_async_to_lds_b128(
      (gv4i*)gsrc, (lv4i*)lds_dst, /*offset=*/0, /*cpol=*/0);
#else
  *(float4*)lds_dst = *(const float4*)gsrc;
#endif
}

__device__ __forceinline__ void gather_fence() {
#if GN_ASYNC
  __builtin_amdgcn_s_wait_asynccnt(0);
#endif
  __syncthreads();
}

// One 16x16 output tile of X(16xK) @ W(KxN-slice at n0), K multiple of 4.
// A-frag (32-bit A 16x4): lane row = lane&15; lanes<16 hold K=k,k+1,
// lanes>=16 hold K=k+2,k+3.  B-frag (4x16): VGPR v, lanes0-15 = row K=k+v,
// lanes16-31 = row K=k+v+2, column = n0 + (lane&15).
// Wp is the packed layout: Wp[((k>>1)*ldw + n)*2 + (k&1)] = W[k][n], so the
// (b.x, b.y) pair for a lane is a single aligned 8-byte load.
__device__ __forceinline__ v8f gemm_tile16(const float* __restrict__ X, int ldx,
                                           const float* __restrict__ Wp, int ldw,
                                           int K, int n0, int lane) {
  v8f acc = {};
  const int hi = lane >> 4;     // 0 or 1
  const int lo = lane & 15;
  const float* xrow = X + lo * ldx + 2 * hi;                  // + k below
  const float* wrow = Wp + ((size_t)hi * ldw + n0 + lo) * 2;  // + k*ldw below
  for (int k = 0; k < K; k += 4) {
    v2f a;
    a.x = xrow[k + 0];
    a.y = xrow[k + 1];
    v2f b = *(const v2f*)(wrow + (size_t)k * ldw);  // global_load_b64
    acc = __builtin_amdgcn_wmma_f32_16x16x4_f32(
        /*neg_a=*/false, a, /*neg_b=*/false, b,
        /*c_mod=*/(short)0, acc, /*reuse_a=*/false, /*reuse_b=*/false);
  }
  return acc;
}

// C/D layout: VGPR i -> row M = i + 8*(lane>=16), col N = n0 + (lane&15).
__device__ __forceinline__ void store_tile_lds(float* __restrict__ H, int ldh,
                                               v8f acc, const float* __restrict__ bias,
                                               int n0, int lane, bool relu) {
  const int hi = lane >> 4;
  const int lo = lane & 15;
  const int col = n0 + lo;
  const float bv = bias[col];
#pragma unroll
  for (int i = 0; i < 8; ++i) {
    float v = acc[i] + bv;
    if (relu) v = fmaxf(v, 0.0f);
    H[(i + 8 * hi) * ldh + col] = v;
  }
}

// ------------------------- weight repack -----------------------------------
// Wp[((k>>1)*Nc + n)*2 + (k&1)] = W[k*Nc + n]
__global__ __launch_bounds__(256) void repack_kernel(const float* __restrict__ W,
                                                     float* __restrict__ Wp,
                                                     int K, int Nc) {
  const int i = blockIdx.x * 256 + threadIdx.x;
  if (i < K * Nc) {
    const int k = i / Nc, n = i % Nc;
    Wp[((size_t)(k >> 1) * Nc + n) * 2 + (k & 1)] = W[i];
  }
}

// ------------------------- edge MLP (e0 then e1) ---------------------------
// ea_out[e] = mlp(mlp(edge_attr[e])), 64 -> 128 -> 64 -> 128 -> 64.
__global__ __launch_bounds__(128) void edge_mlp_kernel(
    const float* __restrict__ ea_in,
    const float* __restrict__ W1a, const float* __restrict__ b1a,
    const float* __restrict__ W2a, const float* __restrict__ b2a,
    const float* __restrict__ W1b, const float* __restrict__ b1b,
    const float* __restrict__ W2b, const float* __restrict__ b2b,
    float* __restrict__ ea_out) {
  constexpr int LDX = GN_DE + 4;    // 68 floats: 16B-aligned rows, bank-skewed
  constexpr int LDH = 128 + 4;      // 132
  __shared__ float X[16 * LDX];
  __shared__ float H[16 * LDH];
  __shared__ float Y[16 * LDX];

  const int tid = threadIdx.x;
  const int wave = tid >> 5;
  const int lane = tid & 31;
  const long e0 = (long)blockIdx.x * 16;

  // Load X tile: 16 x 64 floats = 256 float4, coalesced (async -> LDS).
  for (int idx = tid; idx < 16 * 16; idx += 128) {
    const int e = idx >> 4, c4 = idx & 15;
    copy16_to_lds(&X[e * LDX + c4 * 4], ea_in + (e0 + e) * GN_DE + c4 * 4);
  }
  gather_fence();

  // e0 layer1: 64 -> 128 (relu)
  for (int t = wave; t < 8; t += 4) {
    v8f acc = gemm_tile16(X, LDX, W1a, 128, GN_DE, t * 16, lane);
    store_tile_lds(H, LDH, acc, b1a, t * 16, lane, true);
  }
  __syncthreads();
  // e0 layer2: 128 -> 64
  {
    v8f acc = gemm_tile16(H, LDH, W2a, GN_DE, 128, wave * 16, lane);
    store_tile_lds(Y, LDX, acc, b2a, wave * 16, lane, false);
  }
  __syncthreads();
  // e1 layer1: 64 -> 128 (relu)
  for (int t = wave; t < 8; t += 4) {
    v8f acc = gemm_tile16(Y, LDX, W1b, 128, GN_DE, t * 16, lane);
    store_tile_lds(H, LDH, acc, b1b, t * 16, lane, true);
  }
  __syncthreads();
  // e1 layer2: 128 -> 64, write out
  {
    v8f acc = gemm_tile16(H, LDH, W2b, GN_DE, 128, wave * 16, lane);
    const int hi = lane >> 4, lo = lane & 15;
    const int col = wave * 16 + lo;
    const float bv = b2b[col];
#pragma unroll
    for (int i = 0; i < 8; ++i) {
      const int r = i + 8 * hi;
      ea_out[(e0 + r) * GN_DE + col] = acc[i] + bv;
    }
  }
}

// ------------------------- GNN round (gather + MLP + scatter) --------------
// CIN = 2*128 + ea_dim; MLP CIN -> 256 -> 128; scatter-add into nf_acc[nni].
template <int CIN, int EADIM, bool STORE_EA>
__global__ __launch_bounds__(128) void gnn_layer_kernel(
    const float* __restrict__ nf, const float* __restrict__ ea,
    const int* __restrict__ nodepair, const int* __restrict__ nni,
    const float* __restrict__ W1, const float* __restrict__ b1,
    const float* __restrict__ W2, const float* __restrict__ b2,
    float* __restrict__ ea_out, float* __restrict__ nf_acc) {
  constexpr int CHID = 256, COUT = 128;
  constexpr int LDX = CIN + 4;     // rows stay 16B aligned (CIN % 4 == 0)
  constexpr int LDH = CHID + 4;
  __shared__ float X[16 * LDX];
  __shared__ float H[16 * LDH];

  const int tid = threadIdx.x;
  const int wave = tid >> 5;
  const int lane = tid & 31;
  const long e0 = (long)blockIdx.x * 16;

  // Gather [nf[src] | nf[dst] | ea] into the X tile, 16B chunks, async->LDS.
  constexpr int VEC = CIN / 4;     // 80 or 96; 16*VEC divisible by 128
  for (int idx = tid; idx < 16 * VEC; idx += 128) {
    const int e = idx / VEC;
    const int c = (idx % VEC) * 4;
    const long eg = e0 + e;
    const float* src;
    if (c < 128) {
      src = nf + (size_t)nodepair[eg * 2 + 0] * GN_DN + c;
    } else if (c < 256) {
      src = nf + (size_t)nodepair[eg * 2 + 1] * GN_DN + (c - 128);
    } else {
      src = ea + (size_t)eg * EADIM + (c - 256);
    }
    copy16_to_lds(&X[e * LDX + c], src);
  }
  gather_fence();

  // Layer1: CIN -> 256 (relu); 16 N-tiles over 4 waves.
  for (int t = wave; t < CHID / 16; t += 4) {
    v8f acc = gemm_tile16(X, LDX, W1, CHID, CIN, t * 16, lane);
    store_tile_lds(H, LDH, acc, b1, t * 16, lane, true);
  }
  __syncthreads();

  // Layer2: 256 -> 128; store edge activation + scatter-add segment sum.
  for (int t = wave; t < COUT / 16; t += 4) {
    v8f acc = gemm_tile16(H, LDH, W2, COUT, CHID, t * 16, lane);
    const int hi = lane >> 4, lo = lane & 15;
    const int col = t * 16 + lo;
    const float bv = b2[col];
#pragma unroll
    for (int i = 0; i < 8; ++i) {
      const int r = i + 8 * hi;
      const long eg = e0 + r;
      const float y = acc[i] + bv;
      if constexpr (STORE_EA) ea_out[eg * COUT + col] = y;
      atomicAdd(&nf_acc[(size_t)nni[eg] * COUT + col], y);  // global_atomic_add_f32
    }
  }
}

// ------------------------- small helpers -----------------------------------
__global__ __launch_bounds__(256) void count_kernel(const int* __restrict__ nni,
                                                    int* __restrict__ counts) {
  const int i = blockIdx.x * 256 + threadIdx.x;
  if (i < GN_E) atomicAdd(&counts[nni[i]], 1);
}

__global__ __launch_bounds__(256) void scale_kernel(float* __restrict__ buf,
                                                    const int* __restrict__ counts) {
  const int i = blockIdx.x * 256 + threadIdx.x;
  if (i < GN_N * GN_DN) {
    const int node = i >> 7;  // /128
    const int c = counts[node];
    buf[i] = buf[i] / (float)(c < 1 ? 1 : c);
  }
}

// ---------------------------------------------------------------------------
extern "C" void kernel_launch(void* const* d_in, const int* in_sizes, int n_in,
                              void* d_out, int out_size, void* d_ws, size_t ws_size,
                              hipStream_t stream) {
  (void)in_sizes; (void)n_in; (void)out_size; (void)ws_size;

  const float* node_features = (const float*)d_in[0];   // N x 128
  const float* edge_attr     = (const float*)d_in[1];   // E x 64
  const int*   nodepair      = (const int*)d_in[2];     // E x 2
  const int*   nni           = (const int*)d_in[3];     // E
  const float* e0W1 = (const float*)d_in[4];  const float* e0b1 = (const float*)d_in[5];
  const float* e0W2 = (const float*)d_in[6];  const float* e0b2 = (const float*)d_in[7];
  const float* e1W1 = (const float*)d_in[8];  const float* e1b1 = (const float*)d_in[9];
  const float* e1W2 = (const float*)d_in[10]; const float* e1b2 = (const float*)d_in[11];
  const float* g0W1 = (const float*)d_in[12]; const float* g0b1 = (const float*)d_in[13];
  const float* g0W2 = (const float*)d_in[14]; const float* g0b2 = (const float*)d_in[15];
  const float* g1W1 = (const float*)d_in[16]; const float* g1b1 = (const float*)d_in[17];
  const float* g1W2 = (const float*)d_in[18]; const float* g1b2 = (const float*)d_in[19];

  float* out = (float*)d_out;  // N x 128

  // Workspace carve-out (~322 MB).
  float* ea0 = (float*)d_ws;                        // E x 64
  float* ea1 = ea0 + (size_t)GN_E * GN_DE;          // E x 128
  float* nf1 = ea1 + (size_t)GN_E * 128;            // N x 128
  int* counts = (int*)(nf1 + (size_t)GN_N * GN_DN); // N  (N*4 bytes, 16B-mult)
  float* pk = (float*)(counts + GN_N);              // packed weights (~1.2MB)
  float* pe0W1 = pk;             // 64*128
  float* pe0W2 = pe0W1 + 8192;   // 128*64
  float* pe1W1 = pe0W2 + 8192;
  float* pe1W2 = pe1W1 + 8192;
  float* pg0W1 = pe1W2 + 8192;   // 320*256
  float* pg0W2 = pg0W1 + 81920;  // 256*128
  float* pg1W1 = pg0W2 + 32768;  // 384*256
  float* pg1W2 = pg1W1 + 98304;  // 256*128

  // Repack all weight matrices into B-fragment order (tiny, once per launch).
  auto rp = [&](const float* W, float* Wp, int K, int Nc) {
    repack_kernel<<<(K * Nc + 255) / 256, 256, 0, stream>>>(W, Wp, K, Nc);
  };
  rp(e0W1, pe0W1, 64, 128);  rp(e0W2, pe0W2, 128, 64);
  rp(e1W1, pe1W1, 64, 128);  rp(e1W2, pe1W2, 128, 64);
  rp(g0W1, pg0W1, 320, 256); rp(g0W2, pg0W2, 256, 128);
  rp(g1W1, pg1W1, 384, 256); rp(g1W2, pg1W2, 256, 128);

  const int eb = GN_E / 16;  // 25000 edge tiles (exact)

  // ea = e1(e0(edge_attr))
  edge_mlp_kernel<<<eb, 128, 0, stream>>>(edge_attr,
                                          pe0W1, e0b1, pe0W2, e0b2,
                                          pe1W1, e1b1, pe1W2, e1b2, ea0);

  // neighbour counts / denom
  (void)hipMemsetAsync(counts, 0, (size_t)GN_N * sizeof(int), stream);
  count_kernel<<<(GN_E + 255) / 256, 256, 0, stream>>>(nni, counts);

  // round 0: cin = 128+128+64 = 320
  (void)hipMemsetAsync(nf1, 0, (size_t)GN_N * GN_DN * sizeof(float), stream);
  gnn_layer_kernel<320, 64, true><<<eb, 128, 0, stream>>>(
      node_features, ea0, nodepair, nni, pg0W1, g0b1, pg0W2, g0b2, ea1, nf1);
  scale_kernel<<<(GN_N * GN_DN + 255) / 256, 256, 0, stream>>>(nf1, counts);

  // round 1: cin = 128+128+128 = 384, accumulate straight into d_out
  (void)hipMemsetAsync(out, 0, (size_t)GN_N * GN_DN * sizeof(float), stream);
  gnn_layer_kernel<384, 128, false><<<eb, 128, 0, stream>>>(
      nf1, ea1, nodepair, nni, pg1W1, g1b1, pg1W2, g1b2, nullptr, out);
  scale_kernel<<<(GN_N * GN_DN + 255) / 256, 256, 0, stream>>>(out, counts);
}